// GAT_38113539785175
// MI455X (gfx1250) — compile-verified
//
#include <hip/hip_runtime.h>
#include <hip/hip_bf16.h>
#include <math.h>

// ---------------- problem constants (match reference) ----------------
#define NN    50000     // nodes
#define EIN   800000    // raw edges
#define ETOT  850000    // edges + self loops
#define FDIM  128       // feature width at every layer boundary (H*CH = IN = 128)
#define SLOPE 0.2f
#define BN_EPS 1e-5f

typedef __attribute__((ext_vector_type(2))) float v2f;
typedef __attribute__((ext_vector_type(8))) float v8f;

// ---------------- helpers ----------------
__device__ __forceinline__ void edge_nodes(const int* __restrict__ ei, int e, int& s, int& d) {
    if (e < EIN) { s = ei[e]; d = ei[EIN + e]; }
    else         { s = e - EIN; d = e - EIN; }      // self loops appended
}

// sign-aware float atomic max via int/uint ordering trick
__device__ __forceinline__ void atomicMaxF32(float* addr, float val) {
    int iv = __float_as_int(val);
    if (iv >= 0) atomicMax((int*)addr, iv);
    else         atomicMin((unsigned int*)addr, (unsigned int)iv);
}

// ---------------- 1) GEMM: Hout[N,128] = A[N,128] @ W[128,128] (f32 WMMA) ----------------
// One wave computes a 16-row x 128-col strip: 8 independent 16x16 accumulators,
// K swept in steps of 4 with V_WMMA_F32_16X16X4_F32 (A fragment reused 8x).
__global__ __launch_bounds__(256) void gemm128_wmma(const float* __restrict__ A,
                                                    const float* __restrict__ W,
                                                    float* __restrict__ Hout,
                                                    int nrows) {
    int wave = (int)((blockIdx.x * blockDim.x + threadIdx.x) >> 5);
    int lane = threadIdx.x & 31;
    int m0 = wave * 16;
    if (m0 >= nrows) return;
    int lm = lane & 15;   // row (A) / col (B,C)
    int hi = lane >> 4;   // K-half select (A,B) / M+8 select (C)

    v8f acc[8];
#pragma unroll
    for (int j = 0; j < 8; ++j) acc[j] = (v8f){0.f,0.f,0.f,0.f,0.f,0.f,0.f,0.f};

    const float* arow = A + (size_t)(m0 + lm) * FDIM;
#pragma unroll 4
    for (int k0 = 0; k0 < FDIM; k0 += 4) {
        int ka = k0 + 2 * hi;
        v2f a;
        a.x = arow[ka];
        a.y = arow[ka + 1];
        const float* wr0 = W + (size_t)ka * FDIM + lm;  // row ka, col lm (+16*j)
        const float* wr1 = wr0 + FDIM;                  // row ka+1
#pragma unroll
        for (int j = 0; j < 8; ++j) {
            v2f b;
            b.x = wr0[j * 16];
            b.y = wr1[j * 16];
            acc[j] = __builtin_amdgcn_wmma_f32_16x16x4_f32(
                false, a, false, b, (short)0, acc[j], false, false);
        }
    }
    // C/D layout: VGPR v -> row m0 + v + 8*hi, lane lm -> col j*16 + lm
#pragma unroll
    for (int j = 0; j < 8; ++j) {
        float* orow = Hout + (size_t)(m0 + 8 * hi) * FDIM + j * 16 + lm;
#pragma unroll
        for (int v = 0; v < 8; ++v) orow[(size_t)v * FDIM] = acc[j][v];
    }
}

// ---------------- 2) attention coefficients: al = sum_c h[n,h,c]*a[h,c] ----------------
__global__ __launch_bounds__(256) void attn_coef(const float* __restrict__ Hm,
                                                 const float* __restrict__ a_src,
                                                 const float* __restrict__ a_dst,
                                                 float* __restrict__ als,
                                                 float* __restrict__ ald, int n) {
    int wave = (int)((blockIdx.x * blockDim.x + threadIdx.x) >> 5);
    int lane = threadIdx.x & 31;
    if (wave >= n) return;
    const float* hp = Hm + (size_t)wave * FDIM;
    float h0 = hp[lane], h1 = hp[lane + 32], h2 = hp[lane + 64], h3 = hp[lane + 96];
    float s0 = h0 * a_src[lane]      + h1 * a_src[lane + 32];   // head 0 (c 0..63)
    float s1 = h2 * a_src[lane + 64] + h3 * a_src[lane + 96];   // head 1 (c 64..127)
    float d0 = h0 * a_dst[lane]      + h1 * a_dst[lane + 32];
    float d1 = h2 * a_dst[lane + 64] + h3 * a_dst[lane + 96];
#pragma unroll
    for (int off = 16; off > 0; off >>= 1) {
        s0 += __shfl_xor(s0, off, 32);
        s1 += __shfl_xor(s1, off, 32);
        d0 += __shfl_xor(d0, off, 32);
        d1 += __shfl_xor(d1, off, 32);
    }
    if (lane == 0) {
        als[wave * 2] = s0; als[wave * 2 + 1] = s1;
        ald[wave * 2] = d0; ald[wave * 2 + 1] = d1;
    }
}

// ---------------- 3) per-layer scratch init ----------------
__global__ void init_layer(float* __restrict__ m, float* __restrict__ den,
                           float* __restrict__ acc, float* __restrict__ colsum,
                           float* __restrict__ colsq, int n) {
    int idx = blockIdx.x * blockDim.x + threadIdx.x;
    int stride = gridDim.x * blockDim.x;
    for (int i = idx; i < n * FDIM; i += stride) acc[i] = 0.f;
    for (int i = idx; i < n * 2; i += stride) { m[i] = -__builtin_inff(); den[i] = 0.f; }
    if (idx < FDIM) { colsum[idx] = 0.f; colsq[idx] = 0.f; }
}

// ---------------- 4) edge pass 1: e = leaky(al_s[src]+al_d[dst]); segment max ----------------
__global__ void edge_max_k(const int* __restrict__ ei, const float* __restrict__ als,
                           const float* __restrict__ ald, float* __restrict__ ebuf,
                           float* __restrict__ m) {
    int e = blockIdx.x * blockDim.x + threadIdx.x;
    if (e >= ETOT) return;
    int s, d; edge_nodes(ei, e, s, d);
#pragma unroll
    for (int h = 0; h < 2; ++h) {
        float ev = als[s * 2 + h] + ald[d * 2 + h];
        ev = ev > 0.f ? ev : SLOPE * ev;
        ebuf[e * 2 + h] = ev;
        atomicMaxF32(&m[d * 2 + h], ev);
    }
}

// ---------------- 5) edge pass 2: ex = exp(e - m[dst]); segment sum ----------------
__global__ void edge_exp_k(const int* __restrict__ ei, float* __restrict__ ebuf,
                           const float* __restrict__ m, float* __restrict__ den) {
    int e = blockIdx.x * blockDim.x + threadIdx.x;
    if (e >= ETOT) return;
    int s, d; edge_nodes(ei, e, s, d);
#pragma unroll
    for (int h = 0; h < 2; ++h) {
        float ex = __expf(ebuf[e * 2 + h] - m[d * 2 + h]);
        ebuf[e * 2 + h] = ex;
        atomicAdd(&den[d * 2 + h], ex);
    }
}

// ---------------- 6) edge pass 3: acc[dst] += ex * h[src]  (wave per edge) ----------------
__global__ __launch_bounds__(256) void edge_aggregate(const int* __restrict__ ei,
                                                      const float* __restrict__ ebuf,
                                                      const float* __restrict__ Hm,
                                                      float* __restrict__ acc) {
    int wave = (int)((blockIdx.x * blockDim.x + threadIdx.x) >> 5);
    int lane = threadIdx.x & 31;
    if (wave >= ETOT) return;
    int s, d; edge_nodes(ei, wave, s, d);
    float ex = ebuf[wave * 2 + (lane >> 4)];            // head = (4*lane)>>6
    const float4* hp = (const float4*)(Hm + (size_t)s * FDIM);
    float4 hv = hp[lane];                               // channels 4*lane .. 4*lane+3
    float* ap = acc + (size_t)d * FDIM + lane * 4;
    atomicAdd(ap + 0, hv.x * ex);
    atomicAdd(ap + 1, hv.y * ex);
    atomicAdd(ap + 2, hv.z * ex);
    atomicAdd(ap + 3, hv.w * ex);
}

// ---------------- 7) epilogue (layers 0/1): normalize+bias, accumulate BN col stats ----------------
__global__ __launch_bounds__(256) void epi_bias_stats(float* __restrict__ acc,
                                                      const float* __restrict__ den,
                                                      const float* __restrict__ b,
                                                      float* __restrict__ colsum,
                                                      float* __restrict__ colsq, int n) {
    __shared__ float ls[FDIM];
    __shared__ float lq[FDIM];
    int t = threadIdx.x;
    if (t < FDIM) { ls[t] = 0.f; lq[t] = 0.f; }
    __syncthreads();
    int stride = gridDim.x * blockDim.x;
    for (int i = blockIdx.x * blockDim.x + t; i < n * FDIM; i += stride) {
        int c = i & (FDIM - 1);
        int node = i >> 7;
        int h = c >> 6;
        float v = acc[i] / (den[node * 2 + h] + 1e-16f) + b[c];
        acc[i] = v;                      // in-place: pre-BN features
        atomicAdd(&ls[c], v);
        atomicAdd(&lq[c], v * v);
    }
    __syncthreads();
    if (t < FDIM) { atomicAdd(&colsum[t], ls[t]); atomicAdd(&colsq[t], lq[t]); }
}

// ---------------- 8) BN stats -> per-column scale/shift ----------------
__global__ void bn_stats(const float* __restrict__ colsum, const float* __restrict__ colsq,
                         const float* __restrict__ g, const float* __restrict__ be,
                         float* __restrict__ scale, float* __restrict__ shift) {
    int c = threadIdx.x;
    if (c < FDIM) {
        float mu  = colsum[c] * (1.0f / NN);
        float var = colsq[c] * (1.0f / NN) - mu * mu;
        float rs  = rsqrtf(var + BN_EPS);
        float sc  = rs * g[c];
        scale[c] = sc;
        shift[c] = be[c] - mu * sc;
    }
}

// ---------------- 9) BN apply + ReLU (in place) ----------------
__global__ void bn_apply(float* __restrict__ feat, const float* __restrict__ scale,
                         const float* __restrict__ shift, int n) {
    int stride = gridDim.x * blockDim.x;
    for (int i = blockIdx.x * blockDim.x + threadIdx.x; i < n * FDIM; i += stride) {
        int c = i & (FDIM - 1);
        float v = feat[i] * scale[c] + shift[c];
        feat[i] = v > 0.f ? v : 0.f;
    }
}

// ---------------- 10) final layer: head-mean + b2 ----------------
__global__ void final_out(const float* __restrict__ acc, const float* __restrict__ den,
                          const float* __restrict__ b2, float* __restrict__ out, int n) {
    int stride = gridDim.x * blockDim.x;
    for (int i = blockIdx.x * blockDim.x + threadIdx.x; i < n * 64; i += stride) {
        int node = i >> 6;
        int c = i & 63;
        float v0 = acc[(size_t)node * FDIM + c]      / (den[node * 2]     + 1e-16f);
        float v1 = acc[(size_t)node * FDIM + 64 + c] / (den[node * 2 + 1] + 1e-16f);
        out[i] = 0.5f * (v0 + v1) + b2[c];
    }
}

// ---------------- host driver ----------------
extern "C" void kernel_launch(void* const* d_in, const int* in_sizes, int n_in,
                              void* d_out, int out_size, void* d_ws, size_t ws_size,
                              hipStream_t stream) {
    const float* x    = (const float*)d_in[0];
    const int*   ei   = (const int*)d_in[1];
    const float* W0   = (const float*)d_in[2];
    const float* as0  = (const float*)d_in[3];
    const float* ad0  = (const float*)d_in[4];
    const float* b0   = (const float*)d_in[5];
    const float* g0   = (const float*)d_in[6];
    const float* be0  = (const float*)d_in[7];
    const float* W1   = (const float*)d_in[8];
    const float* as1  = (const float*)d_in[9];
    const float* ad1  = (const float*)d_in[10];
    const float* b1   = (const float*)d_in[11];
    const float* g1   = (const float*)d_in[12];
    const float* be1  = (const float*)d_in[13];
    const float* W2   = (const float*)d_in[14];
    const float* as2  = (const float*)d_in[15];
    const float* ad2  = (const float*)d_in[16];
    const float* b2   = (const float*)d_in[17];
    float* out = (float*)d_out;

    // workspace layout (floats)
    float* ws = (float*)d_ws;
    const size_t NF = (size_t)NN * FDIM;
    float* B0     = ws;                 // rotating N x 128 buffers
    float* B1     = B0 + NF;
    float* B2     = B1 + NF;
    float* als    = B2 + NF;            // N x 2
    float* ald    = als + (size_t)NN * 2;
    float* mbuf   = ald + (size_t)NN * 2;
    float* den    = mbuf + (size_t)NN * 2;
    float* ebuf   = den + (size_t)NN * 2;          // ETOT x 2
    float* colsum = ebuf + (size_t)ETOT * 2;
    float* colsq  = colsum + FDIM;
    float* scale  = colsq + FDIM;
    float* shift  = scale + FDIM;

    const int gemm_waves  = NN / 16;                         // 3125 (N % 16 == 0)
    const int gemm_grid   = (gemm_waves + 7) / 8;
    const int attn_grid   = (NN + 7) / 8;
    const int edge_grid   = (ETOT + 255) / 256;
    const int aggr_grid   = (ETOT + 7) / 8;                  // wave per edge

    // ---------------- layer 0: x -> B0(h) -> B1(acc -> feat) ----------------
    gemm128_wmma<<<gemm_grid, 256, 0, stream>>>(x, W0, B0, NN);
    attn_coef<<<attn_grid, 256, 0, stream>>>(B0, as0, ad0, als, ald, NN);
    init_layer<<<2048, 256, 0, stream>>>(mbuf, den, B1, colsum, colsq, NN);
    edge_max_k<<<edge_grid, 256, 0, stream>>>(ei, als, ald, ebuf, mbuf);
    edge_exp_k<<<edge_grid, 256, 0, stream>>>(ei, ebuf, mbuf, den);
    edge_aggregate<<<aggr_grid, 256, 0, stream>>>(ei, ebuf, B0, B1);
    epi_bias_stats<<<1024, 256, 0, stream>>>(B1, den, b0, colsum, colsq, NN);
    bn_stats<<<1, 128, 0, stream>>>(colsum, colsq, g0, be0, scale, shift);
    bn_apply<<<2048, 256, 0, stream>>>(B1, scale, shift, NN);

    // ---------------- layer 1: B1 -> B0(h) -> B2(acc -> feat) ----------------
    gemm128_wmma<<<gemm_grid, 256, 0, stream>>>(B1, W1, B0, NN);
    attn_coef<<<attn_grid, 256, 0, stream>>>(B0, as1, ad1, als, ald, NN);
    init_layer<<<2048, 256, 0, stream>>>(mbuf, den, B2, colsum, colsq, NN);
    edge_max_k<<<edge_grid, 256, 0, stream>>>(ei, als, ald, ebuf, mbuf);
    edge_exp_k<<<edge_grid, 256, 0, stream>>>(ei, ebuf, mbuf, den);
    edge_aggregate<<<aggr_grid, 256, 0, stream>>>(ei, ebuf, B0, B2);
    epi_bias_stats<<<1024, 256, 0, stream>>>(B2, den, b1, colsum, colsq, NN);
    bn_stats<<<1, 128, 0, stream>>>(colsum, colsq, g1, be1, scale, shift);
    bn_apply<<<2048, 256, 0, stream>>>(B2, scale, shift, NN);

    // ---------------- layer 2: B2 -> B0(h) -> B1(acc) -> out ----------------
    gemm128_wmma<<<gemm_grid, 256, 0, stream>>>(B2, W2, B0, NN);
    attn_coef<<<attn_grid, 256, 0, stream>>>(B0, as2, ad2, als, ald, NN);
    init_layer<<<2048, 256, 0, stream>>>(mbuf, den, B1, colsum, colsq, NN);
    edge_max_k<<<edge_grid, 256, 0, stream>>>(ei, als, ald, ebuf, mbuf);
    edge_exp_k<<<edge_grid, 256, 0, stream>>>(ei, ebuf, mbuf, den);
    edge_aggregate<<<aggr_grid, 256, 0, stream>>>(ei, ebuf, B0, B1);
    final_out<<<2048, 256, 0, stream>>>(B1, den, b2, out, NN);
}